// PQC_22308060135906
// MI455X (gfx1250) — compile-verified
//
#include <hip/hip_runtime.h>
#include <math.h>

#define N_WIRES  14
#define DIM      16384          // 2^14 amplitudes
#define QDEPTH   12
#define NTHREADS 512            // 16 waves of 32

typedef float v2f __attribute__((ext_vector_type(2)));
typedef float v8f __attribute__((ext_vector_type(8)));
typedef unsigned int u32x4 __attribute__((ext_vector_type(4)));
typedef int          i32x4 __attribute__((ext_vector_type(4)));
typedef int          i32x8 __attribute__((ext_vector_type(8)));

// Composed CNOT ring layer: src-index permutation (linear over GF(2)).
// x -> sigma_0(sigma_1(...sigma_13(x))) with sigma_q: bit_t ^= bit_c, c=q, t=(q+r)%14.
__device__ __forceinline__ int cnot_perm(int x, int r) {
    #pragma unroll
    for (int q = N_WIRES - 1; q >= 0; --q) {
        int t  = q + r; if (t >= N_WIRES) t -= N_WIRES;
        int cb = (N_WIRES - 1) - q;       // wire -> index bit
        int tb = (N_WIRES - 1) - t;
        x ^= ((x >> cb) & 1) << tb;
    }
    return x;
}

// One workgroup simulates one batch element; state (Re/Im planes) stays
// resident in LDS across all 12 layers.  Rot-gate layers are fused into
// four 16x16 complex unitaries applied with V_WMMA_F32_16X16X4_F32.
// Input row is staged global->LDS with the Tensor Data Mover when available.
__global__ __launch_bounds__(NTHREADS, 1)
void pqc_sim_kernel(const float* __restrict__ inp,
                    const float* __restrict__ weights,
                    float* __restrict__ out)
{
    __shared__ float sRe[DIM];          // 64 KB
    __shared__ float sIm[DIM];          // 64 KB
    __shared__ float sUr [4][256];      // group unitaries: Re
    __shared__ float sUi [4][256];      //                  Im
    __shared__ float sUin[4][256];      //                  -Im (f32 WMMA has no A-neg)
    __shared__ float g2r[16][4];        // per-wire 2x2 Rot matrices (Re), [14]=identity
    __shared__ float g2i[16][4];
    __shared__ float red[NTHREADS];

    const int b    = blockIdx.x;
    const int tid  = threadIdx.x;
    const int wave = tid >> 5;
    const int lane = tid & 31;
    const int half = lane >> 4;   // K/M half select per WMMA layout
    const int l16  = lane & 15;

    // ---------- stage input row into LDS ----------
    for (int i = tid; i < DIM; i += NTHREADS) sIm[i] = 0.f;

#if __has_builtin(__builtin_amdgcn_tensor_load_to_lds)
    if (wave == 0) {
        // Tensor DMA descriptor: 1-D tile, 16384 x f32, global -> sRe.
        const unsigned long long ga =
            (unsigned long long)(const void*)(inp + (size_t)b * DIM);
        const unsigned int lds_addr = (unsigned int)(size_t)(void*)sRe; // low 32b = LDS offset
        u32x4 g0 = { 1u,                                    // count=1 (valid user D#)
                     lds_addr,                              // lds_addr
                     (unsigned int)(ga & 0xffffffffu),      // global_addr[31:0]
                     (unsigned int)((ga >> 32) & 0x01ffffffu) | (2u << 30) }; // addr[56:32], type=2
        i32x8 g1 = { (int)(2u << 16),       // data_size=4B, wg_mask=0, no pad/iterate
                     (int)(0x4000u << 16),  // tensor_dim0[15:0]=16384 @ bits[63:48]
                     (int)(1u << 16),       // tensor_dim0[31:16]=0, tensor_dim1[15:0]=1
                     (int)(0x4000u << 16),  // tensor_dim1[31:16]=0, tile_dim0=16384
                     0,                     // tile_dim1=0 (unused), tile_dim2=0
                     16384,                 // tensor_dim0_stride[31:0]
                     0, 0 };                // stride hi / tensor_dim1_stride
        i32x4 z4 = { 0, 0, 0, 0 };
#if __clang_major__ >= 23
        i32x8 z8 = { 0, 0, 0, 0, 0, 0, 0, 0 };
        __builtin_amdgcn_tensor_load_to_lds(g0, g1, z4, z4, z8, 0);
#else
        __builtin_amdgcn_tensor_load_to_lds(g0, g1, z4, z4, 0);
#endif
        __builtin_amdgcn_s_wait_tensorcnt(0);
    }
    __syncthreads();
    float acc = 0.f;
    for (int i = tid; i < DIM; i += NTHREADS) {
        float v = sRe[i];
        acc += v * v;
    }
#else
    float acc = 0.f;
    for (int i = tid; i < DIM; i += NTHREADS) {
        float v = inp[(size_t)b * DIM + i];
        sRe[i] = v;
        acc += v * v;
    }
#endif

    // ---------- block reduction for the L2 norm ----------
    red[tid] = acc;
    __syncthreads();
    for (int s = NTHREADS / 2; s > 0; s >>= 1) {
        if (tid < s) red[tid] += red[tid + s];
        __syncthreads();
    }
    const float inv = 1.0f / sqrtf(red[0]);   // folded into layer-0/group-0 unitary

    const int lotab[4] = {10, 6, 2, 0};   // LSB of each 4-bit wire group in the index

    for (int l = 0; l < QDEPTH; ++l) {
        // ---------- per-wire 2x2 Rot matrices ----------
        if (tid < N_WIRES) {
            const float* wp = weights + ((size_t)l * N_WIRES + tid) * 3;
            float phi = tanhf(wp[0]);
            float th  = tanhf(wp[1]);
            float om  = tanhf(wp[2]);
            float c = cosf(0.5f * th), s = sinf(0.5f * th);
            float apo = 0.5f * (phi + om), amo = 0.5f * (phi - om);
            float ca = cosf(apo), sa = sinf(apo);
            float cm = cosf(amo), sm = sinf(amo);
            g2r[tid][0] =  ca * c;  g2i[tid][0] = -sa * c;   // e^{-i(phi+om)/2} c
            g2r[tid][1] = -cm * s;  g2i[tid][1] = -sm * s;   // -e^{+i(phi-om)/2} s
            g2r[tid][2] =  cm * s;  g2i[tid][2] = -sm * s;   // e^{-i(phi-om)/2} s
            g2r[tid][3] =  ca * c;  g2i[tid][3] =  sa * c;   // e^{+i(phi+om)/2} c
        } else if (tid == N_WIRES) {      // virtual identity wire
            g2r[14][0] = 1.f; g2r[14][1] = 0.f; g2r[14][2] = 0.f; g2r[14][3] = 1.f;
            g2i[14][0] = 0.f; g2i[14][1] = 0.f; g2i[14][2] = 0.f; g2i[14][3] = 0.f;
        }
        __syncthreads();

        // ---------- Kron-product 16x16 unitaries (4 groups x 256 elems) ----------
        for (int e = tid; e < 1024; e += NTHREADS) {
            int g  = e >> 8;
            int ik = e & 255;
            int i  = ik >> 4, k = ik & 15;
            float re = 1.f, im = 0.f;
            for (int p = 3; p >= 0; --p) {
                // wire occupying index-bit (lo+p); wire q <-> bit (13-q)
                int w = (g < 3) ? (4 * g + 3 - p)
                                : ((p == 0) ? 13 : (p == 1) ? 12 : 14);
                int ai = (i >> p) & 1, bi = (k >> p) & 1;
                float er = g2r[w][ai * 2 + bi], ei = g2i[w][ai * 2 + bi];
                float nr = re * er - im * ei;
                im = re * ei + im * er;
                re = nr;
            }
            if (l == 0 && g == 0) { re *= inv; im *= inv; }  // amplitude normalization
            sUr [g][ik] = re;
            sUi [g][ik] = im;
            sUin[g][ik] = -im;
        }
        __syncthreads();

        // ---------- apply the four 16x16 complex unitaries with WMMA ----------
        for (int g = 0; g < 4; ++g) {
            const int lo     = lotab[g];
            const int lomask = (1 << lo) - 1;

            // A fragments of U (shared by all 4 tiles of this wave)
            v2f Ar[4], Ai[4], Ain[4];
            #pragma unroll
            for (int c = 0; c < 4; ++c) {
                int k    = 4 * c + 2 * half;
                int base = l16 * 16 + k;           // row = lane%16, K-half by lane>=16
                Ar [c].x = sUr [g][base]; Ar [c].y = sUr [g][base + 1];
                Ai [c].x = sUi [g][base]; Ai [c].y = sUi [g][base + 1];
                Ain[c].x = sUin[g][base]; Ain[c].y = sUin[g][base + 1];
            }

            #pragma unroll
            for (int tt = 0; tt < 4; ++tt) {
                int t = wave * 4 + tt;             // 64 column-tiles over 16 waves
                int m = t * 16 + l16;              // this lane's column
                int colbase = ((m >> lo) << (lo + 4)) + (m & lomask);

                v2f Br[4], Bi[4];
                #pragma unroll
                for (int c = 0; c < 4; ++c) {
                    int k  = 4 * c + 2 * half;
                    int a0 = colbase + (k << lo);
                    int a1 = a0 + (1 << lo);
                    Br[c].x = sRe[a0]; Br[c].y = sRe[a1];
                    Bi[c].x = sIm[a0]; Bi[c].y = sIm[a1];
                }

                v8f dR = {0.f,0.f,0.f,0.f,0.f,0.f,0.f,0.f};
                v8f dI = {0.f,0.f,0.f,0.f,0.f,0.f,0.f,0.f};
                // Re = Ur*Sr + (-Ui)*Si ; Im = Ur*Si + Ui*Sr
                #pragma unroll
                for (int c = 0; c < 4; ++c)
                    dR = __builtin_amdgcn_wmma_f32_16x16x4_f32(
                             false, Ar[c], false, Br[c], (short)0, dR, false, false);
                #pragma unroll
                for (int c = 0; c < 4; ++c)
                    dR = __builtin_amdgcn_wmma_f32_16x16x4_f32(
                             false, Ain[c], false, Bi[c], (short)0, dR, false, false);
                #pragma unroll
                for (int c = 0; c < 4; ++c)
                    dI = __builtin_amdgcn_wmma_f32_16x16x4_f32(
                             false, Ar[c], false, Bi[c], (short)0, dI, false, false);
                #pragma unroll
                for (int c = 0; c < 4; ++c)
                    dI = __builtin_amdgcn_wmma_f32_16x16x4_f32(
                             false, Ai[c], false, Br[c], (short)0, dI, false, false);

                // D layout: VGPR v -> row v + 8*half, col = lane%16
                #pragma unroll
                for (int v = 0; v < 8; ++v) {
                    int i    = v + 8 * half;
                    int addr = colbase + (i << lo);
                    sRe[addr] = dR[v];
                    sIm[addr] = dI[v];
                }
            }
            __syncthreads();
        }

        // ---------- CNOT ring layer as one GF(2)-linear permutation ----------
        // Linearity: src(tid ^ (u<<9)) = perm(tid) ^ perm(u<<9).  The five
        // u-bit masks are wave-uniform (SALU); per element it is one v_xor.
        const int r = (l % (N_WIRES - 1)) + 1;
        const int base = cnot_perm(tid, r);          // per-thread (VALU)
        const int m9   = cnot_perm(1 << 9,  r);      // uniform -> scalar regs
        const int m10  = cnot_perm(1 << 10, r);
        const int m11  = cnot_perm(1 << 11, r);
        const int m12  = cnot_perm(1 << 12, r);
        const int m13  = cnot_perm(1 << 13, r);

        float tr[32], ti[32];
        #pragma unroll
        for (int u = 0; u < 32; ++u) {
            int xo = ((u & 1)  ? m9  : 0) ^ ((u & 2)  ? m10 : 0) ^
                     ((u & 4)  ? m11 : 0) ^ ((u & 8)  ? m12 : 0) ^
                     ((u & 16) ? m13 : 0);
            int x = base ^ xo;
            tr[u] = sRe[x];
            ti[u] = sIm[x];
        }
        __syncthreads();                             // all reads before any write
        #pragma unroll
        for (int u = 0; u < 32; ++u) {
            int j = tid + (u << 9);
            sRe[j] = tr[u];
            sIm[j] = ti[u];
        }
        __syncthreads();
    }

    // ---------- probs = |amp|^2 ----------
    for (int i = tid; i < DIM; i += NTHREADS) {
        float re = sRe[i], im = sIm[i];
        out[(size_t)b * DIM + i] = re * re + im * im;
    }
}

extern "C" void kernel_launch(void* const* d_in, const int* in_sizes, int n_in,
                              void* d_out, int out_size, void* d_ws, size_t ws_size,
                              hipStream_t stream) {
    (void)n_in; (void)out_size; (void)d_ws; (void)ws_size;
    const float* inp = (const float*)d_in[0];   // (64, 16384) f32
    const float* w   = (const float*)d_in[1];   // (12, 14, 3) f32
    float* out = (float*)d_out;                 // (64, 16384) f32
    const int batch = in_sizes[0] / DIM;
    pqc_sim_kernel<<<batch, NTHREADS, 0, stream>>>(inp, w, out);
}